// ResidualGNNs_34806414966790
// MI455X (gfx1250) — compile-verified
//
#include <hip/hip_runtime.h>
#include <math.h>

// ---------------- problem constants (match reference) ----------------
#define F_NODES   360
#define NUM_GRAPHS 128
#define CCH       128                         // hidden_channels
#define NNODES    (NUM_GRAPHS * F_NODES)      // 46080
#define INPUT_DIM 64980                       // F*(F+1)/2
#define ZCOLS     (INPUT_DIM + 3 * CCH)       // 65364
#define HID       512
#define HID2      256
#define EPSB      1e-5f
#define KB        16                          // k-values staged per LDS chunk

typedef float v2f __attribute__((ext_vector_type(2)));
typedef float v8f __attribute__((ext_vector_type(8)));

// ---------------- tiny utility kernels ----------------
__global__ __launch_bounds__(256) void k_fill0(float* p, int n) {
  int i = blockIdx.x * 256 + threadIdx.x;
  if (i < n) p[i] = 0.f;
}

__global__ __launch_bounds__(256) void k_deg_init(float* deg, int n) {
  int i = blockIdx.x * 256 + threadIdx.x;
  if (i < n) deg[i] = 1.f;                    // self-loop contributes 1
}

__global__ __launch_bounds__(256) void k_deg_accum(const int* __restrict__ dst,
                                                   float* deg, int E) {
  int e = blockIdx.x * 256 + threadIdx.x;
  if (e < E) atomicAdd(&deg[dst[e]], 1.f);
}

__global__ __launch_bounds__(256) void k_dinv(float* deg, int n) {
  int i = blockIdx.x * 256 + threadIdx.x;
  if (i < n) deg[i] = rsqrtf(deg[i]);         // deg >= 1 always
}

// ---------------- fp32 WMMA GEMM: C[M,N] = A[M,K] @ B[K,N] ----------------
// Block = 256 threads = 8 waves stacked along M -> block tile 128(M) x 64(N).
// Each wave owns a 16x64 strip (4 accumulators reuse one A fragment).
// The 16x64 B tile is staged in LDS once per chunk, k-pair interleaved:
//   element (k, n) -> Bs[(k/2)*128 + 2n + (k&1)]
// so a B fragment (B[k,n], B[k+1,n]) is a single aligned ds_load_b64 that
// lands directly in the even VGPR pair the WMMA wants (no repack movs).
// All 8 waves share the tile. K tail (K % 16, multiple of 4) is consumed
// from global memory by the last split-K slice.
// Requirements: K % 4 == 0; M % 128 == 0 or single 128 block; N % 64 == 0.
// gridDim.z > 1 => split-K with f32 atomic accumulation (C pre-zeroed).
__global__ __launch_bounds__(256) void k_gemm_wmma_f32(
    const float* __restrict__ A, const float* __restrict__ B,
    float* __restrict__ Cmat, int M, int N, int K, int cps) {
  __shared__ float Bs[(KB / 2) * 128];        // 1024 floats = 4 KB
  const int tid  = threadIdx.x;
  const int lane = tid & 31;
  const int wave = tid >> 5;
  const int half = lane >> 4;                 // 0: K {0,1}, 1: K {2,3}
  const int mrow = lane & 15;
  const size_t m0 = (size_t)blockIdx.x * 128 + (size_t)wave * 16;
  const int n0 = blockIdx.y * 64;

  const int chunk_tot = K / KB;               // full 16-k chunks
  int c0 = blockIdx.z * cps;
  int c1 = c0 + cps; if (c1 > chunk_tot) c1 = chunk_tot;

  v8f acc[4] = {};
  const float* arow = A + (m0 + (size_t)mrow) * (size_t)K;

  // B staging: thread covers k-row pair kp (0..7) and 2 columns (col2, col2+1)
  const int kp   = tid >> 5;                  // 0..7
  const int col2 = (tid & 31) << 1;           // 0,2,...,62

  for (int ch = c0; ch < c1; ++ch) {
    const int kbase = ch * KB;
    __syncthreads();
    {
      const float* b0 = B + (size_t)(kbase + 2 * kp) * (size_t)N + n0 + col2;
      const float2 e = *(const float2*)b0;        // row k = 2*kp
      const float2 o = *(const float2*)(b0 + N);  // row k = 2*kp + 1
      *(float4*)(Bs + kp * 128 + col2 * 2) = make_float4(e.x, o.x, e.y, o.y);
    }
    __syncthreads();
    if (kbase + KB < K)                       // prefetch next A chunk
      __builtin_prefetch(arow + kbase + KB, 0, 0);
#pragma unroll
    for (int ks = 0; ks < KB / 4; ++ks) {
      const int k0 = kbase + ks * 4 + 2 * half;
      v2f a; a.x = arow[k0]; a.y = arow[k0 + 1];
      const float* lb = Bs + (ks * 2 + half) * 128 + mrow * 2;
#pragma unroll
      for (int t = 0; t < 4; ++t) {
        const v2f b = *(const v2f*)(lb + t * 32);   // one ds_load_b64
        acc[t] = __builtin_amdgcn_wmma_f32_16x16x4_f32(
            false, a, false, b, (short)0, acc[t], false, false);
      }
    }
  }

  // K tail (K % KB, multiple of 4): last split-K slice only, direct global B
  if (blockIdx.z == gridDim.z - 1) {
    for (int k = chunk_tot * KB; k < K; k += 4) {
      const int k0 = k + 2 * half;
      v2f a; a.x = arow[k0]; a.y = arow[k0 + 1];
      const float* bp = B + (size_t)k0 * (size_t)N + n0 + mrow;
#pragma unroll
      for (int t = 0; t < 4; ++t) {
        v2f b; b.x = bp[t * 16]; b.y = bp[N + t * 16];
        acc[t] = __builtin_amdgcn_wmma_f32_16x16x4_f32(
            false, a, false, b, (short)0, acc[t], false, false);
      }
    }
  }

  // epilogue: C/D layout VGPR j -> M = j + 8*half, N = lane%16 (+16 per acc)
  if (gridDim.z > 1) {
#pragma unroll
    for (int t = 0; t < 4; ++t)
#pragma unroll
      for (int j = 0; j < 8; ++j)
        atomicAdd(Cmat + (m0 + (size_t)(j + 8 * half)) * (size_t)N
                       + n0 + t * 16 + mrow, acc[t][j]);
  } else {
#pragma unroll
    for (int t = 0; t < 4; ++t)
#pragma unroll
      for (int j = 0; j < 8; ++j)
        Cmat[(m0 + (size_t)(j + 8 * half)) * (size_t)N
             + n0 + t * 16 + mrow] = acc[t][j];
  }
}

// ---------------- GCN aggregation ----------------
// out[i,:] = dinv[i]^2 * tmp[i,:]   (self-loop term, also initializes out)
__global__ __launch_bounds__(256) void k_scatter_init(
    const float* __restrict__ tmp, const float* __restrict__ dinv,
    float* __restrict__ outb, int n) {
  int idx = blockIdx.x * 256 + threadIdx.x;
  if (idx >= n) return;
  int node = idx >> 7;                        // /128 channels
  float w = dinv[node];
  outb[idx] = tmp[idx] * w * w;
}

// one edge per wave; lane covers 4 channels (float4) -> 128 channels
__global__ __launch_bounds__(256) void k_scatter_edges(
    const float* __restrict__ tmp, const float* __restrict__ dinv,
    const int* __restrict__ src, const int* __restrict__ dst,
    float* __restrict__ outb, int E) {
  int e = blockIdx.x * 8 + (threadIdx.x >> 5);
  if (e >= E) return;
  int lane = threadIdx.x & 31;
  int s = src[e], d = dst[e];
  float norm = dinv[s] * dinv[d];
  const float4 v = ((const float4*)(tmp + (size_t)s * CCH))[lane];
  float* po = outb + (size_t)d * CCH + lane * 4;
  atomicAdd(po + 0, v.x * norm);
  atomicAdd(po + 1, v.y * norm);
  atomicAdd(po + 2, v.z * norm);
  atomicAdd(po + 3, v.w * norm);
}

// in-place: h = tanh(agg + bias)
__global__ __launch_bounds__(256) void k_gcn_finish(
    float* agg, const float* __restrict__ bias, int n) {
  int idx = blockIdx.x * 256 + threadIdx.x;
  if (idx < n) agg[idx] = tanhf(agg[idx] + bias[idx & (CCH - 1)]);
}

// per-graph mean pool of one layer's features into z
// grid = NUM_GRAPHS blocks, 128 threads (one per channel)
__global__ __launch_bounds__(128) void k_pool(
    const float* __restrict__ h, float* __restrict__ z, int layer) {
  int g = blockIdx.x, c = threadIdx.x;
  const float* base = h + (size_t)g * F_NODES * CCH + c;
  float s = 0.f;
  for (int i = 0; i < F_NODES; ++i) s += base[(size_t)i * CCH];
  z[(size_t)g * ZCOLS + INPUT_DIM + layer * CCH + c] = s * (1.0f / F_NODES);
}

// upper-triangular gather of raw connectome: grid (B, F), 256 threads
__global__ __launch_bounds__(256) void k_triu(
    const float* __restrict__ x, float* __restrict__ z) {
  int g = blockIdx.x, i = blockIdx.y;
  int off = i * F_NODES - (i * (i - 1)) / 2;  // start of row i in triu order
  const float* xr = x + ((size_t)g * F_NODES + i) * F_NODES;
  float* zr = z + (size_t)g * ZCOLS + off - i;
  for (int j = i + threadIdx.x; j < F_NODES; j += 256) zr[j] = xr[j];
}

// BatchNorm (training stats) over the 128-graph batch for all ZCOLS columns,
// selecting (bn_g,bn_b) for triu part and (bnh_g,bnh_b) for pooled part.
__global__ __launch_bounds__(256) void k_bn_z(
    float* __restrict__ z,
    const float* __restrict__ bn_g, const float* __restrict__ bn_b,
    const float* __restrict__ bnh_g, const float* __restrict__ bnh_b) {
  int col = blockIdx.x * 256 + threadIdx.x;
  if (col >= ZCOLS) return;
  float s = 0.f, s2 = 0.f;
  for (int r = 0; r < NUM_GRAPHS; ++r) {
    float v = z[(size_t)r * ZCOLS + col];
    s += v; s2 += v * v;
  }
  float m = s * (1.0f / NUM_GRAPHS);
  float var = s2 * (1.0f / NUM_GRAPHS) - m * m;   // biased, matches jnp.var
  float rs = rsqrtf(var + EPSB);
  float gg = (col < INPUT_DIM) ? bn_g[col] : bnh_g[col - INPUT_DIM];
  float bb = (col < INPUT_DIM) ? bn_b[col] : bnh_b[col - INPUT_DIM];
  for (int r = 0; r < NUM_GRAPHS; ++r) {
    size_t p = (size_t)r * ZCOLS + col;
    z[p] = (z[p] - m) * rs * gg + bb;
  }
}

// BN (training stats) + ReLU, in place, over 128 rows x `cols` columns.
// NOTE: linear bias before BN cancels exactly (mean-subtraction), so skipped.
__global__ __launch_bounds__(256) void k_bn_relu(
    float* __restrict__ a, const float* __restrict__ g,
    const float* __restrict__ be, int cols) {
  int col = blockIdx.x * 256 + threadIdx.x;
  if (col >= cols) return;
  float s = 0.f, s2 = 0.f;
  for (int r = 0; r < NUM_GRAPHS; ++r) {
    float v = a[(size_t)r * cols + col];
    s += v; s2 += v * v;
  }
  float m = s * (1.0f / NUM_GRAPHS);
  float var = s2 * (1.0f / NUM_GRAPHS) - m * m;
  float rs = rsqrtf(var + EPSB);
  float gg = g[col], bb = be[col];
  for (int r = 0; r < NUM_GRAPHS; ++r) {
    size_t p = (size_t)r * cols + col;
    float v = (a[p] - m) * rs * gg + bb;
    a[p] = v > 0.f ? v : 0.f;
  }
}

// final linear (256 -> 2) + log_softmax; one thread per graph
__global__ __launch_bounds__(128) void k_head(
    const float* __restrict__ a3, const float* __restrict__ W4,
    const float* __restrict__ b4, float* __restrict__ out) {
  int g = threadIdx.x;
  if (g >= NUM_GRAPHS) return;
  const float* row = a3 + (size_t)g * HID2;
  float l0 = b4[0], l1 = b4[1];
  for (int k = 0; k < HID2; ++k) {
    float v = row[k];
    l0 += v * W4[k * 2 + 0];
    l1 += v * W4[k * 2 + 1];
  }
  float mx = fmaxf(l0, l1);
  float lse = mx + logf(expf(l0 - mx) + expf(l1 - mx));
  out[g * 2 + 0] = l0 - lse;
  out[g * 2 + 1] = l1 - lse;
}

// ---------------- launcher ----------------
extern "C" void kernel_launch(void* const* d_in, const int* in_sizes, int n_in,
                              void* d_out, int out_size, void* d_ws, size_t ws_size,
                              hipStream_t stream) {
  const float* x    = (const float*)d_in[0];
  const int* esrc   = (const int*)d_in[1];
  const int* edst   = (const int*)d_in[2];
  // d_in[3] = batch (unused; graphs are contiguous blocks of F_NODES rows)
  const float* Wc1  = (const float*)d_in[4];
  const float* bc1  = (const float*)d_in[5];
  const float* Wc2  = (const float*)d_in[6];
  const float* bc2  = (const float*)d_in[7];
  const float* Wc3  = (const float*)d_in[8];
  const float* bc3  = (const float*)d_in[9];
  const float* bn_g = (const float*)d_in[10];
  const float* bn_b = (const float*)d_in[11];
  const float* bnh_g= (const float*)d_in[12];
  const float* bnh_b= (const float*)d_in[13];
  const float* W1   = (const float*)d_in[14];
  const float* g1   = (const float*)d_in[16];
  const float* be1  = (const float*)d_in[17];
  const float* W2   = (const float*)d_in[18];
  const float* g2   = (const float*)d_in[20];
  const float* be2  = (const float*)d_in[21];
  const float* W3   = (const float*)d_in[22];
  const float* g3   = (const float*)d_in[24];
  const float* be3  = (const float*)d_in[25];
  const float* W4   = (const float*)d_in[26];
  const float* b4   = (const float*)d_in[27];
  float* out        = (float*)d_out;

  const int E = in_sizes[1];
  const int NB = NNODES * CCH;                 // 5,898,240 floats per node buffer

  // workspace carving (floats)
  float* ws   = (float*)d_ws;
  float* B0   = ws;
  float* B1   = B0 + (size_t)NB;
  float* B2   = B1 + (size_t)NB;
  float* z    = B2 + (size_t)NB;               // 128 x 65364
  float* dinv = z + (size_t)NUM_GRAPHS * ZCOLS;
  float* a1   = dinv + NNODES;                 // 128 x 512
  float* a2   = a1 + NUM_GRAPHS * HID;         // 128 x 256
  float* a3   = a2 + NUM_GRAPHS * HID2;        // 128 x 256

  dim3 blk(256);
  const int gN   = (NNODES + 255) / 256;
  const int gNB  = (NB + 255) / 256;
  const int gE   = (E + 255) / 256;
  const int gEw  = (E + 7) / 8;                // one edge per wave

  // --- degrees -> D^{-1/2} ---
  k_deg_init<<<gN, blk, 0, stream>>>(dinv, NNODES);
  k_deg_accum<<<gE, blk, 0, stream>>>(edst, dinv, E);
  k_dinv<<<gN, blk, 0, stream>>>(dinv, NNODES);

  // --- GCN layer 1: x[46080x360] @ Wc1[360x128] (22 chunks + 8-k tail) ---
  k_gemm_wmma_f32<<<dim3(NNODES / 128, CCH / 64, 1), blk, 0, stream>>>(
      x, Wc1, B0, NNODES, CCH, F_NODES, F_NODES / KB);
  k_scatter_init<<<gNB, blk, 0, stream>>>(B0, dinv, B1, NB);
  k_scatter_edges<<<gEw, blk, 0, stream>>>(B0, dinv, esrc, edst, B1, E);
  k_gcn_finish<<<gNB, blk, 0, stream>>>(B1, bc1, NB);
  k_pool<<<NUM_GRAPHS, 128, 0, stream>>>(B1, z, 0);

  // --- GCN layer 2: h1 @ Wc2[128x128] ---
  k_gemm_wmma_f32<<<dim3(NNODES / 128, CCH / 64, 1), blk, 0, stream>>>(
      B1, Wc2, B2, NNODES, CCH, CCH, CCH / KB);
  k_scatter_init<<<gNB, blk, 0, stream>>>(B2, dinv, B0, NB);
  k_scatter_edges<<<gEw, blk, 0, stream>>>(B2, dinv, esrc, edst, B0, E);
  k_gcn_finish<<<gNB, blk, 0, stream>>>(B0, bc2, NB);
  k_pool<<<NUM_GRAPHS, 128, 0, stream>>>(B0, z, 1);

  // --- GCN layer 3: h2 @ Wc3[128x128] ---
  k_gemm_wmma_f32<<<dim3(NNODES / 128, CCH / 64, 1), blk, 0, stream>>>(
      B0, Wc3, B2, NNODES, CCH, CCH, CCH / KB);
  k_scatter_init<<<gNB, blk, 0, stream>>>(B2, dinv, B1, NB);
  k_scatter_edges<<<gEw, blk, 0, stream>>>(B2, dinv, esrc, edst, B1, E);
  k_gcn_finish<<<gNB, blk, 0, stream>>>(B1, bc3, NB);
  k_pool<<<NUM_GRAPHS, 128, 0, stream>>>(B1, z, 2);

  // --- build z: triu gather of raw x, then BN over batch ---
  k_triu<<<dim3(NUM_GRAPHS, F_NODES), blk, 0, stream>>>(x, z);
  k_bn_z<<<(ZCOLS + 255) / 256, blk, 0, stream>>>(z, bn_g, bn_b, bnh_g, bnh_b);

  // --- MLP layer 1: z[128x65364] @ W1[65364x512], split-K=128 ---
  k_fill0<<<(NUM_GRAPHS * HID + 255) / 256, blk, 0, stream>>>(a1, NUM_GRAPHS * HID);
  {
    int chunk_tot = ZCOLS / KB;                // 4085 (+ one 4-k tail step)
    int SK = 128, cps = (chunk_tot + SK - 1) / SK;  // 32
    k_gemm_wmma_f32<<<dim3(1, HID / 64, SK), blk, 0, stream>>>(
        z, W1, a1, NUM_GRAPHS, HID, ZCOLS, cps);
  }
  k_bn_relu<<<(HID + 255) / 256, blk, 0, stream>>>(a1, g1, be1, HID);

  // --- MLP layer 2: a1[128x512] @ W2[512x256], split-K=16 ---
  k_fill0<<<(NUM_GRAPHS * HID2 + 255) / 256, blk, 0, stream>>>(a2, NUM_GRAPHS * HID2);
  k_gemm_wmma_f32<<<dim3(1, HID2 / 64, 16), blk, 0, stream>>>(
      a1, W2, a2, NUM_GRAPHS, HID2, HID, (HID / KB) / 16);
  k_bn_relu<<<(HID2 + 255) / 256, blk, 0, stream>>>(a2, g2, be2, HID2);

  // --- MLP layer 3: a2[128x256] @ W3[256x256], split-K=8 ---
  k_fill0<<<(NUM_GRAPHS * HID2 + 255) / 256, blk, 0, stream>>>(a3, NUM_GRAPHS * HID2);
  k_gemm_wmma_f32<<<dim3(1, HID2 / 64, 8), blk, 0, stream>>>(
      a2, W3, a3, NUM_GRAPHS, HID2, HID2, (HID2 / KB) / 8);
  k_bn_relu<<<(HID2 + 255) / 256, blk, 0, stream>>>(a3, g3, be3, HID2);

  // --- head: 256 -> 2 + log_softmax ---
  k_head<<<1, 128, 0, stream>>>(a3, W4, b4, out);
}